// MultiBoxLoss_36928128811174
// MI455X (gfx1250) — compile-verified
//
#include <hip/hip_runtime.h>
#include <stdint.h>

// ---------------------------------------------------------------------------
// MultiBoxLoss for MI455X (gfx1250).
//   B=64 images, P=65536 anchors, C=2 classes, NEGPOS=7.
// Memory-bound problem: dense traffic ~587MB, but smooth-L1 terms only matter
// for ~2% positive anchors -> predicated loads cut it to ~120MB (~5us at
// 23.3 TB/s).  Hard-negative mining (double argsort -> rank<k) is replaced by
// an exact per-image radix top-k select: sum over mined negatives =
// sum(loss_c > v_k) + ties_needed * v_k  (index tie-breaks never change a sum
// over equal values).  loss_c rows are staged through the 320KB LDS using the
// CDNA5 Tensor Data Mover (tensor_load_to_lds) and async global->LDS copies.
//
// Workspace layout (d_ws):  [0..3] f32 loss accumulators,
//   u32[4..6] element counts, u32[12..75] per-image num_pos,
//   float[256 ...] loss_c[B*P]  (needs ~16.8 MB + 1KB of scratch).
// ---------------------------------------------------------------------------

#define BB 64
#define PP 65536
#define NEGPOS 7

typedef unsigned int u32x4 __attribute__((ext_vector_type(4)));
typedef unsigned int u32x8 __attribute__((ext_vector_type(8)));

// ---------------- CDNA5 (gfx1250) data-movement helpers --------------------

__device__ __forceinline__ uint32_t lds_addr_of(const void* p) {
  // generic pointer into LDS aperture: low 32 bits == workgroup LDS byte offset
  return (uint32_t)(uintptr_t)p;
}

// Per-lane async copy: 16B global -> LDS, tracked by ASYNCcnt.
__device__ __forceinline__ void async_load_b128(uint32_t lds, const void* g) {
  asm volatile("global_load_async_to_lds_b128 %0, %1, off"
               :: "v"(lds), "v"((uint64_t)(uintptr_t)g)
               : "memory");
}
__device__ __forceinline__ void wait_asynccnt(int n) {
  if (n) asm volatile("s_wait_asynccnt 1" ::: "memory");
  else   asm volatile("s_wait_asynccnt 0" ::: "memory");
}

// TDM: contiguous 1-D f32 tile (n < 65536 elems), global -> LDS.
// D# per CDNA5 ISA: group0 (count=1, lds_addr,
// global_addr, type=2) + group1 (data_size=4B, dims/strides), groups 2/3 NULL.
__device__ __forceinline__ void tdm_load_f32(uint32_t lds, const void* g, uint32_t n) {
  uint64_t ga = (uint64_t)(uintptr_t)g;
  u32x4 g0; u32x8 g1;
  g0[0] = 1u;                                        // count=1, user mode
  g0[1] = lds;                                       // lds_addr (bytes)
  g0[2] = (uint32_t)ga;                              // global_addr[31:0]
  g0[3] = ((uint32_t)(ga >> 32) & 0x01FFFFFFu)       // global_addr[56:32]
          | (2u << 30);                              // type=2 ("image")
  g1[0] = 2u << 16;                                  // data_size=4B; no mask/pad
  g1[1] = (n & 0xFFFFu) << 16;                       // tensor_dim0[15:0]
  g1[2] = (n >> 16) | (1u << 16);                    // tensor_dim0[31:16] | tensor_dim1=1
  g1[3] = (n & 0xFFFFu) << 16;                       // tile_dim0 = n
  g1[4] = 1u;                                        // tile_dim1=1, tile_dim2 unused
  g1[5] = n;                                         // tensor_dim0_stride[31:0]
  g1[6] = (n & 0xFFFFu) << 16;                       // stride0_hi=0 | stride1[15:0]
  g1[7] = n >> 16;                                   // stride1 hi
  asm volatile("tensor_load_to_lds %0, %1" :: "s"(g0), "s"(g1) : "memory");
}
__device__ __forceinline__ void wait_tensorcnt(int n) {
  if (n) __builtin_amdgcn_s_wait_tensorcnt(1);
  else   __builtin_amdgcn_s_wait_tensorcnt(0);
}

// ---------------------------- kernels --------------------------------------

__global__ void k_init(unsigned* ws) { ws[threadIdx.x] = 0u; }  // 256 threads

__device__ __forceinline__ void sl1(float t, float p, float& s, unsigned& c) {
  if (t == t) {                 // ~isnan(target); pred NaN propagates like ref
    float ad = fabsf(t - p);
    s += (ad < 1.0f) ? 0.5f * ad * ad : ad - 0.5f;
    c++;
  }
}

__global__ __launch_bounds__(256) void k_stream(
    const float* __restrict__ boxes_pred, const float* __restrict__ conf_pred,
    const float* __restrict__ kpts_pred,  const float* __restrict__ dpth_pred,
    const float* __restrict__ boxes_t,    const float* __restrict__ kypts_t,
    const float* __restrict__ dpths_t,    const int* __restrict__ label_t,
    float* __restrict__ acc, unsigned* __restrict__ cnts,
    float* __restrict__ loss_c)
{
  const int b = blockIdx.y;
  const size_t idx = (size_t)b * PP + (size_t)blockIdx.x * 256 + threadIdx.x;

  const bool pos = label_t[idx] != 0;

  float sL = 0.f, sC = 0.f, sK = 0.f, sD = 0.f;
  unsigned cL = 0, cK = 0, cD = 0;

  if (pos) {   // smooth-L1 terms only exist for positive anchors (~2%)
    float4 t4 = ((const float4*)boxes_t)[idx];
    float4 p4 = ((const float4*)boxes_pred)[idx];
    sl1(t4.x, p4.x, sL, cL); sl1(t4.y, p4.y, sL, cL);
    sl1(t4.z, p4.z, sL, cL); sl1(t4.w, p4.w, sL, cL);

    const float2* kt = (const float2*)(kypts_t  + idx * 10);
    const float2* kp = (const float2*)(kpts_pred + idx * 10);
#pragma unroll
    for (int i = 0; i < 5; ++i) {
      float2 t2 = kt[i], p2 = kp[i];
      sl1(t2.x, p2.x, sK, cK); sl1(t2.y, p2.y, sK, cK);
    }
    float2 dt = ((const float2*)dpths_t)[idx];
    float2 dp = ((const float2*)dpth_pred)[idx];
    sl1(dt.x, dp.x, sD, cD); sl1(dt.y, dp.y, sD, cD);
  }

  // mining score / CE pieces (label in {0,1}; binarized label == label_t)
  float2 cf = ((const float2*)conf_pred)[idx];
  float mx  = fmaxf(cf.x, cf.y);
  float lse = mx + logf(expf(cf.x - mx) + expf(cf.y - mx));
  float lc;
  if (pos) { sC = lse - cf.y; lc = 0.0f; }   // positives: CE counted now, score 0
  else     { lc = lse - cf.x; }              // negatives: score == their CE
  loss_c[idx] = lc;

  // block reduction: 4 float sums + 4 small counts (exact in f32)
  float v[8] = { sL, sC, sK, sD, (float)cL, (float)cK, (float)cD, pos ? 1.f : 0.f };
#pragma unroll
  for (int off = 16; off > 0; off >>= 1)
#pragma unroll
    for (int i = 0; i < 8; ++i) v[i] += __shfl_down(v[i], off, 32);

  __shared__ float red[8][8];
  const int wv = threadIdx.x >> 5;
  if ((threadIdx.x & 31) == 0)
#pragma unroll
    for (int i = 0; i < 8; ++i) red[wv][i] = v[i];
  __syncthreads();
  if (threadIdx.x == 0) {
    float t[8] = {0, 0, 0, 0, 0, 0, 0, 0};
    for (int w = 0; w < 8; ++w)
      for (int i = 0; i < 8; ++i) t[i] += red[w][i];
    atomicAdd(&acc[0], t[0]);                       // loss_l
    atomicAdd(&acc[1], t[1]);                       // loss_conf (positive part)
    atomicAdd(&acc[2], t[2]);                       // loss_landm
    atomicAdd(&acc[3], t[3]);                       // loss_dpth
    atomicAdd(&cnts[0], (unsigned)(t[4] + 0.5f));   // nl
    atomicAdd(&cnts[1], (unsigned)(t[5] + 0.5f));   // n1
    atomicAdd(&cnts[2], (unsigned)(t[6] + 0.5f));   // ndpth
    atomicAdd(&cnts[8 + b], (unsigned)(t[7] + 0.5f)); // num_pos[b]
  }
}

// One block per image: exact k-th-largest via 4x8-bit MSB radix select
// (loss_c >= 0, so float bits order as u32), then sum of mined negatives.
__global__ __launch_bounds__(1024) void k_select(
    const float* __restrict__ loss_c, const unsigned* __restrict__ cnts,
    float* __restrict__ acc)
{
  __shared__ float    rowbuf[2 * 4096];   // 32KB double-buffered LDS stage
  __shared__ unsigned hist[256];
  __shared__ unsigned sscan[256];
  __shared__ float    sred[32];
  __shared__ unsigned s_prefix, s_remk, s_selbin, s_selabove;

  const int b   = blockIdx.x;
  const int tid = threadIdx.x;

  const unsigned np = cnts[8 + b];
  unsigned k = NEGPOS * np;
  if (k > PP - 1) k = PP - 1;
  if (k == 0) return;                      // no mined negatives for this image

  const float*   row   = loss_c + (size_t)b * PP;
  const uint32_t lbase = lds_addr_of(rowbuf);

  if (tid == 0) { s_prefix = 0u; s_remk = k; }
  __syncthreads();

  for (int d = 0; d < 4; ++d) {
    const int      shift  = 24 - 8 * d;
    const unsigned hmask  = (d == 0) ? 0u : (0xFFFFFFFFu << (32 - 8 * d));
    const unsigned prefix = s_prefix;
    if (tid < 256) hist[tid] = 0u;

    // double-buffered TDM staging: wave 0 drives 16x 16KB DMA tiles
    if (tid < 32) tdm_load_f32(lbase, row, 4096);
    for (int c = 0; c < 16; ++c) {
      if (tid < 32) {
        if (c < 15)
          tdm_load_f32(lbase + (unsigned)((c + 1) & 1) * 16384u,
                       row + (size_t)(c + 1) * 4096, 4096);
        wait_tensorcnt(c < 15 ? 1 : 0);    // in-order: tile c complete
      }
      __syncthreads();
      const float* buf = rowbuf + (c & 1) * 4096;
#pragma unroll
      for (int j = 0; j < 4; ++j) {
        unsigned key = __float_as_uint(buf[tid + j * 1024]);
        if ((key & hmask) == prefix)
          atomicAdd(&hist[(key >> shift) & 255u], 1u);
      }
      __syncthreads();
    }

    // suffix-inclusive scan of the 256 bins (Hillis-Steele, high->low)
    if (tid < 256) sscan[tid] = hist[tid];
    __syncthreads();
    for (int off = 1; off < 256; off <<= 1) {
      unsigned x = 0;
      if (tid < 256) { x = sscan[tid]; if (tid + off < 256) x += sscan[tid + off]; }
      __syncthreads();
      if (tid < 256) sscan[tid] = x;
      __syncthreads();
    }
    const unsigned remk = s_remk;
    if (tid < 256) {
      unsigned cum   = sscan[tid];
      unsigned above = (tid < 255) ? sscan[tid + 1] : 0u;
      if (cum >= remk && above < remk) { s_selbin = (unsigned)tid; s_selabove = above; }
    }
    __syncthreads();
    if (tid == 0) {
      s_prefix |= s_selbin << shift;
      s_remk   -= s_selabove;
    }
    __syncthreads();
  }

  const unsigned vbits = s_prefix;   // exact bit pattern of the k-th largest
  const unsigned ties  = s_remk;     // how many elements equal to it are mined

  // sum of mined negatives: stage row via per-lane async global->LDS b128 copies
  float ssum = 0.0f;
  async_load_b128(lbase + (unsigned)tid * 16u, row + (size_t)tid * 4);
  for (int c = 0; c < 16; ++c) {
    if (c < 15)
      async_load_b128(lbase + (unsigned)((c + 1) & 1) * 16384u + (unsigned)tid * 16u,
                      row + (size_t)(c + 1) * 4096 + (size_t)tid * 4);
    wait_asynccnt(c < 15 ? 1 : 0);
    __syncthreads();
    const float* buf = rowbuf + (c & 1) * 4096;
#pragma unroll
    for (int j = 0; j < 4; ++j) {
      float x = buf[tid + j * 1024];
      if (__float_as_uint(x) > vbits) ssum += x;
    }
    __syncthreads();
  }

#pragma unroll
  for (int off = 16; off > 0; off >>= 1) ssum += __shfl_down(ssum, off, 32);
  if ((tid & 31) == 0) sred[tid >> 5] = ssum;
  __syncthreads();
  if (tid == 0) {
    float tot = 0.0f;
    for (int w = 0; w < 32; ++w) tot += sred[w];
    tot += (float)ties * __uint_as_float(vbits);
    atomicAdd(&acc[1], tot);
  }
}

__global__ void k_final(const float* __restrict__ acc,
                        const unsigned* __restrict__ cnts,
                        float* __restrict__ out)
{
  if (threadIdx.x == 0) {
    unsigned tp = 0;
    for (int b = 0; b < BB; ++b) tp += cnts[8 + b];
    float n = fmaxf((float)tp, 1.0f);
    out[0] = acc[0] / fmaxf((float)cnts[0], 1.0f);  // loss_l   / nl
    out[1] = acc[1] / n;                             // loss_conf/ n
    out[2] = acc[2] / fmaxf((float)cnts[1], 1.0f);  // landm    / n1
    out[3] = acc[3] / fmaxf((float)cnts[2], 1.0f);  // dpth     / ndpth
  }
}

// ---------------------------- launcher -------------------------------------

extern "C" void kernel_launch(void* const* d_in, const int* in_sizes, int n_in,
                              void* d_out, int out_size, void* d_ws, size_t ws_size,
                              hipStream_t stream) {
  (void)in_sizes; (void)n_in; (void)out_size; (void)ws_size;
  const float* boxes_pred = (const float*)d_in[0];
  const float* conf_pred  = (const float*)d_in[1];
  const float* kpts_pred  = (const float*)d_in[2];
  const float* dpth_pred  = (const float*)d_in[3];
  const float* boxes_t    = (const float*)d_in[4];
  const float* kypts_t    = (const float*)d_in[5];
  const float* dpths_t    = (const float*)d_in[6];
  const int*   label_t    = (const int*)d_in[7];

  float*    acc    = (float*)d_ws;          // 4 loss accumulators
  unsigned* cnts   = (unsigned*)d_ws + 4;   // [0..2] counts, [8..71] num_pos
  float*    loss_c = (float*)d_ws + 256;    // B*P mining scores (16.8 MB)

  k_init<<<1, 256, 0, stream>>>((unsigned*)d_ws);
  k_stream<<<dim3(PP / 256, BB), 256, 0, stream>>>(
      boxes_pred, conf_pred, kpts_pred, dpth_pred,
      boxes_t, kypts_t, dpths_t, label_t, acc, cnts, loss_c);
  k_select<<<BB, 1024, 0, stream>>>(loss_c, cnts, acc);
  k_final<<<1, 64, 0, stream>>>(acc, cnts, (float*)d_out);
}